// DynamicInputChannels_14688788152638
// MI455X (gfx1250) — compile-verified
//
#include <hip/hip_runtime.h>

typedef float v2f __attribute__((ext_vector_type(2)));
typedef float v8f __attribute__((ext_vector_type(8)));

#define NUM_ROBOTS 16
#define IN_DIM     1024
#define OUT_DIM    1024
#define BATCH      2048
#define NCOLS      (NUM_ROBOTS * OUT_DIM)   // 16384 output columns

#define BM 64          // batch rows per workgroup
#define BN 64          // output cols per workgroup (divides 1024 -> never straddles robots)
#define BK 64          // K chunk staged in LDS
#define LDT 68         // padded LDS row stride in floats (272 B: 16B-aligned, bank-spread)
#define THREADS 128    // 4 waves of 32

__global__ __launch_bounds__(THREADS)
void robots_gemm_wmma_f32(const float* __restrict__ x,
                          const float* __restrict__ W,
                          const float* __restrict__ bias,
                          float* __restrict__ out) {
  // A tile: natural row-major (A-fragment pairs K{2h,2h+1} are contiguous).
  // B tile: stored with per-4-group swizzle {k0,k2,k1,k3} so B-fragment pairs
  //         K{h,h+2} are contiguous -> both operands read with ds_load_b64,
  //         zero per-read selects, conflict-free across all 64 banks.
  __shared__ __align__(16) float lds_a[BM * LDT];
  __shared__ __align__(16) float lds_b[BN * LDT];

  const int t     = threadIdx.x;
  const int lane  = t & 31;
  const int wave  = t >> 5;        // 0..3
  const int wr    = wave >> 1;     // wave's 32-row slab (0/1)
  const int wc    = wave & 1;      // wave's 32-col slab (0/1)
  const int l15   = lane & 15;
  const int lhalf = lane >> 4;     // 0: lanes 0-15, 1: lanes 16-31

  const int col0 = blockIdx.x * BN;   // global output column base
  const int m0   = blockIdx.y * BM;   // global batch-row base

  const int r  = col0 >> 10;             // robot id (BN=64 divides 1024)
  const int o0 = col0 & (OUT_DIM - 1);   // output-channel base within robot
  const float* __restrict__ Wr = W + (size_t)r * OUT_DIM * IN_DIM;

  v8f acc[2][2] = {};   // four 16x16 f32 accumulators per wave

  const int KCHUNKS = IN_DIM / BK;   // 16
  for (int kc = 0; kc < KCHUNKS; ++kc) {
    const int k0 = kc * BK;

    // ---- cooperative tile load: 64 rows x 16 float4 each for A and B ----
    #pragma unroll
    for (int i = 0; i < 8; ++i) {
      const int idx = t + i * THREADS;  // 0..1023
      const int row = idx >> 4;         // 0..63
      const int cv  = idx & 15;         // float4 slot 0..15
      const float4 va = *(const float4*)(x  + (size_t)(m0 + row) * IN_DIM + k0 + cv * 4);
      *(float4*)&lds_a[row * LDT + cv * 4] = va;
      const float4 vb = *(const float4*)(Wr + (size_t)(o0 + row) * IN_DIM + k0 + cv * 4);
      float4 sb;                         // swizzle {k0,k1,k2,k3} -> {k0,k2,k1,k3}
      sb.x = vb.x; sb.y = vb.z; sb.z = vb.y; sb.w = vb.w;
      *(float4*)&lds_b[row * LDT + cv * 4] = sb;
    }

    // Prefetch next chunk: 128 threads x 1 cacheline covers the full 64x64 tile.
    if (kc + 1 < KCHUNKS) {
      const int k1   = k0 + BK;
      const int prow = t >> 1;          // 0..63
      const int pcol = (t & 1) * 32;    // two 128B cachelines per row
      __builtin_prefetch(x  + (size_t)(m0 + prow) * IN_DIM + k1 + pcol, 0, 0);
      __builtin_prefetch(Wr + (size_t)(o0 + prow) * IN_DIM + k1 + pcol, 0, 0);
    }

    __syncthreads();

    // ---- compute: 16 K-steps of 4, four v_wmma_f32_16x16x4_f32 per step ----
    #pragma unroll
    for (int ks = 0; ks < BK; ks += 4) {
      v2f afrag[2], bfrag[2];

      #pragma unroll
      for (int mi = 0; mi < 2; ++mi) {
        // A 16x4: lane (h,m) holds K{2h,2h+1} -> contiguous b64 read
        afrag[mi] = *(const v2f*)&lds_a[(wr * 32 + mi * 16 + l15) * LDT
                                        + ks + 2 * lhalf];
      }
      #pragma unroll
      for (int ni = 0; ni < 2; ++ni) {
        // B 4x16: lane (h,n) holds K{h,h+2} -> contiguous b64 read (swizzled store)
        bfrag[ni] = *(const v2f*)&lds_b[(wc * 32 + ni * 16 + l15) * LDT
                                        + ks + 2 * lhalf];
      }

      #pragma unroll
      for (int mi = 0; mi < 2; ++mi)
        #pragma unroll
        for (int ni = 0; ni < 2; ++ni)
          acc[mi][ni] = __builtin_amdgcn_wmma_f32_16x16x4_f32(
              /*neg_a=*/false, afrag[mi],
              /*neg_b=*/false, bfrag[ni],
              /*c_mod=*/(short)0, acc[mi][ni],
              /*reuse_a=*/false, /*reuse_b=*/false);
    }

    __syncthreads();
  }

  // ---- epilogue: bias add + store ----
  // D layout: VGPR j, lanes 0-15 -> (M=j, N=l15); lanes 16-31 -> (M=j+8, N=l15)
  #pragma unroll
  for (int mi = 0; mi < 2; ++mi) {
    #pragma unroll
    for (int ni = 0; ni < 2; ++ni) {
      const int col = col0 + wc * 32 + ni * 16 + l15;
      const float bv = bias[col];               // flat [16,1024] index == column
      const int rowbase = m0 + wr * 32 + mi * 16 + lhalf * 8;
      #pragma unroll
      for (int j = 0; j < 8; ++j) {
        out[(size_t)(rowbase + j) * NCOLS + col] = acc[mi][ni][j] + bv;
      }
    }
  }
}

extern "C" void kernel_launch(void* const* d_in, const int* in_sizes, int n_in,
                              void* d_out, int out_size, void* d_ws, size_t ws_size,
                              hipStream_t stream) {
  const float* x    = (const float*)d_in[0];   // [2048, 1024]
  const float* W    = (const float*)d_in[1];   // [16, 1024, 1024]
  const float* bias = (const float*)d_in[2];   // [16, 1024]
  float* out = (float*)d_out;                  // [2048, 16384]

  dim3 grid(NCOLS / BN, BATCH / BM);           // (256, 32)
  dim3 block(THREADS);
  robots_gemm_wmma_f32<<<grid, block, 0, stream>>>(x, W, bias, out);
}